// InformationPreservingNorm2D_45483703664819
// MI455X (gfx1250) — compile-verified
//
#include <hip/hip_runtime.h>
#include <hip/hip_bf16.h>
#include <math.h>

// Problem constants (fixed by the reference):
//   x: (160, 64, 64, 64) fp32 ; out: (160, 66, 64, 64) fp32
//   16 groups of 10 samples; per-group scalar mean / biased std over 2,621,440 elems
#define EPSV        1e-5f
#define N_GROUPS    16
#define N_PER_GROUP 2621440      // 10*64*64*64
#define TOTAL_IN    41943040     // 160*64*64*64
#define BLK_PER_GRP 256
#define NBLK1       (N_GROUPS * BLK_PER_GRP)          // 4096
#define CHUNK       (N_PER_GROUP / BLK_PER_GRP)       // 10240 floats per block
#define ITER1       (CHUNK / (256 * 4))               // 10 iterations of 4 floats per thread
#define NORM_BLOCKS (160 * 66)                        // one (b,c) plane per block

typedef __attribute__((ext_vector_type(2))) float v2f;
typedef __attribute__((ext_vector_type(4))) float v4f;
typedef __attribute__((ext_vector_type(8))) float v8f;

#if defined(__has_builtin)
#if __has_builtin(__builtin_amdgcn_wmma_f32_16x16x4_f32)
#define HAVE_WMMA_F32 1
#endif
#endif

// D = A(16x4 f32) * B(4x16 f32, all ones) + C  -> every column of D holds row-sums.
// Total of the 64-element tile = (sum of 8 D regs, lanes 0-15) + (same, lanes 16-31).
__device__ __forceinline__ v8f wmma_rowsum_acc(v2f a, v2f ones, v8f c) {
#ifdef HAVE_WMMA_F32
    return __builtin_amdgcn_wmma_f32_16x16x4_f32(
        /*neg_a=*/false, a, /*neg_b=*/false, ones,
        /*c_mod=*/(short)0, c, /*reuse_a=*/false, /*reuse_b=*/false);
#else
    // Fallback (should not trigger on gfx1250 per probe): plain VALU accumulate.
    c[0] += a[0] + a[1];
    return c;
#endif
}

// ---------------------------------------------------------------------------
// Pass 1: per-block partial (sum, sum_sq) over a contiguous 10,240-float chunk.
// EXEC is all-ones throughout the WMMA loop (exact grid/block tiling, no bounds).
// ---------------------------------------------------------------------------
__global__ __launch_bounds__(256) void ipn_reduce_kernel(
    const float* __restrict__ x, float* __restrict__ part) {
    const int tid = threadIdx.x;
    const v4f* __restrict__ xp = (const v4f*)(x + (long)blockIdx.x * CHUNK);

    const v2f ones = {1.0f, 1.0f};
    v8f accS = {};  // accumulates sums (row-sum columns)
    v8f accQ = {};  // accumulates sums of squares

#pragma unroll
    for (int i = 0; i < ITER1; ++i) {
        v4f v = xp[i * 256 + tid];
        v2f a0 = {v.x, v.y};
        v2f a1 = {v.z, v.w};
        v2f s01 = a0 + a1;              // packed f32 add
        v2f q01 = a0 * a0 + a1 * a1;    // packed mul + fma
        accS = wmma_rowsum_acc(s01, ones, accS);
        accQ = wmma_rowsum_acc(q01, ones, accQ);
    }

    // Per-lane: sum the 8 D registers (M=0..7 half or M=8..15 half of row-sums).
    float s = accS[0] + accS[1] + accS[2] + accS[3] +
              accS[4] + accS[5] + accS[6] + accS[7];
    float q = accQ[0] + accQ[1] + accQ[2] + accQ[3] +
              accQ[4] + accQ[5] + accQ[6] + accQ[7];
    // Combine the two M-halves across the wave32 (lane n <-> lane n+16).
    s += __shfl_xor(s, 16, 32);
    q += __shfl_xor(q, 16, 32);

    __shared__ float ssum[8];
    __shared__ float ssq[8];
    const int wid  = tid >> 5;
    const int lane = tid & 31;
    if (lane == 0) { ssum[wid] = s; ssq[wid] = q; }
    __syncthreads();
    if (tid == 0) {
        float S = 0.f, Q = 0.f;
#pragma unroll
        for (int i = 0; i < 8; ++i) { S += ssum[i]; Q += ssq[i]; }
        part[(long)blockIdx.x * 2 + 0] = S;
        part[(long)blockIdx.x * 2 + 1] = Q;
    }
}

// ---------------------------------------------------------------------------
// Pass 2: fold 256 partials per group into (mean, std).
// ---------------------------------------------------------------------------
__global__ __launch_bounds__(256) void ipn_finalize_kernel(
    const float* __restrict__ part, float* __restrict__ stats) {
    const int g = blockIdx.x;
    const int t = threadIdx.x;
    float S = part[(long)(g * BLK_PER_GRP + t) * 2 + 0];
    float Q = part[(long)(g * BLK_PER_GRP + t) * 2 + 1];
#pragma unroll
    for (int o = 16; o > 0; o >>= 1) {
        S += __shfl_down(S, o, 32);
        Q += __shfl_down(Q, o, 32);
    }
    __shared__ float sS[8];
    __shared__ float sQ[8];
    if ((t & 31) == 0) { sS[t >> 5] = S; sQ[t >> 5] = Q; }
    __syncthreads();
    if (t == 0) {
        float St = 0.f, Qt = 0.f;
#pragma unroll
        for (int i = 0; i < 8; ++i) { St += sS[i]; Qt += sQ[i]; }
        const float invN = 1.0f / (float)N_PER_GROUP;
        float mean = St * invN;
        float var  = fmaxf(Qt * invN - mean * mean, 0.0f);  // biased (ddof=0)
        stats[g * 2 + 0] = mean;
        stats[g * 2 + 1] = sqrtf(var);
    }
}

// ---------------------------------------------------------------------------
// Pass 3: normalize + scale/bias; channels 64/65 are broadcast mean/std.
// One (b, c) plane of 4096 floats per block. x re-read likely hits L2 (192 MB);
// output stored non-temporally (write-once, never re-read).
// ---------------------------------------------------------------------------
__global__ __launch_bounds__(256) void ipn_normalize_kernel(
    const float* __restrict__ x, const float* __restrict__ stats,
    const float* __restrict__ w, const float* __restrict__ bia,
    float* __restrict__ out) {
    const int blk = blockIdx.x;
    const int b = blk / 66;
    const int c = blk % 66;
    const int g = b / 10;

    const float mean = stats[g * 2 + 0];
    const float stdv = stats[g * 2 + 1];
    const float wc = w[c];
    const float bc = bia[c];

    const int tid = threadIdx.x;
    v4f* __restrict__ op = (v4f*)(out + ((long)(b * 66 + c) << 12));

    if (c < 64) {
        const v4f* __restrict__ ip = (const v4f*)(x + ((long)(b * 64 + c) << 12));
        const float s = wc / (stdv + EPSV);
        const float t = bc - mean * s;
#pragma unroll
        for (int i = 0; i < 4; ++i) {
            v4f v = ip[i * 256 + tid];
            v4f r;
            r.x = fmaf(v.x, s, t);
            r.y = fmaf(v.y, s, t);
            r.z = fmaf(v.z, s, t);
            r.w = fmaf(v.w, s, t);
            __builtin_nontemporal_store(r, &op[i * 256 + tid]);
        }
    } else {
        const float val = fmaf((c == 64) ? mean : stdv, wc, bc);
        v4f r = {val, val, val, val};
#pragma unroll
        for (int i = 0; i < 4; ++i) {
            __builtin_nontemporal_store(r, &op[i * 256 + tid]);
        }
    }
}

extern "C" void kernel_launch(void* const* d_in, const int* in_sizes, int n_in,
                              void* d_out, int out_size, void* d_ws, size_t ws_size,
                              hipStream_t stream) {
    const float* x   = (const float*)d_in[0];  // (160,64,64,64) f32
    const float* w   = (const float*)d_in[1];  // (66,) f32
    const float* bia = (const float*)d_in[2];  // (66,) f32
    float* out = (float*)d_out;                // (160,66,64,64) f32

    float* part  = (float*)d_ws;               // NBLK1*2 floats of partials
    float* stats = part + (size_t)NBLK1 * 2;   // 16*2 floats: (mean, std) per group

    ipn_reduce_kernel<<<NBLK1, 256, 0, stream>>>(x, part);
    ipn_finalize_kernel<<<N_GROUPS, 256, 0, stream>>>(part, stats);
    ipn_normalize_kernel<<<NORM_BLOCKS, 256, 0, stream>>>(x, stats, w, bia, out);
}